// FourierAttn2_5970004542054
// MI455X (gfx1250) — compile-verified
//
#include <hip/hip_runtime.h>
#include <hip/hip_bf16.h>

// Problem constants
#define N_HEAD  8
#define D_HEAD  16
#define D_MODEL 128
#define LEN_Q   256
#define LEN_K   512   // MEM_LEN + LEN_Q
#define BATCH   4

typedef __attribute__((ext_vector_type(2))) float v2f;
typedef __attribute__((ext_vector_type(8))) float v8f;

// ---------------------------------------------------------------------------
// Phase 1: fused Q/K projection GEMM using V_WMMA_F32_16X16X4_F32.
// One wave (32 threads) computes one 16x16 tile of (A @ W^T), scaled by
// paramR[head] (the /pi cancels against sinc's internal pi).
//
// Grid: x = row-tile (0..63 -> Q proj over h[1024x128]);
//           (64..191 -> K proj over concat(mems,h)[2048x128])
//       y = feature-tile (0..7), 16 features each.
//
// Fragment layouts per CDNA5 ISA 7.12.2:
//   A 16x4 f32 : lanes 0-15 hold row M=lane,   K={0,1} in v0,v1
//                lanes 16-31 hold row M=lane-16, K={2,3} in v0,v1
//   B 4x16 f32 : lanes 0-15 hold col N=lane,   K={0,1} in v0,v1
//                lanes 16-31 hold col N=lane-16, K={2,3} in v0,v1
//   C/D 16x16  : VGPR j: lanes 0-15 -> (M=j,   N=lane)
//                        lanes 16-31-> (M=j+8, N=lane-16)
// ---------------------------------------------------------------------------
__global__ __launch_bounds__(32)
void proj_wmma(const float* __restrict__ hmat,   // [1024 x 128] = [Lq*B, Dm]
               const float* __restrict__ mems,   // [1024 x 128] = [M*B, Dm]
               const float* __restrict__ Wq,     // [128 x 128] (row = out feat)
               const float* __restrict__ Wk,     // [128 x 128]
               const float* __restrict__ paramR, // [8]
               float* __restrict__ Qs,           // [H][B][Lq][16]
               float* __restrict__ Ks)           // [H][B][Lk][16]
{
    const int lane  = threadIdx.x;          // 0..31
    const int lm    = lane & 15;
    const int khalf = (lane >> 4) << 1;     // 0 for lanes 0-15, 2 for 16-31

    const int  rowTile = blockIdx.x;        // 0..191
    const int  nBase   = blockIdx.y << 4;   // feature base (0..112)
    const bool isK     = rowTile >= 64;

    const float* W   = isK ? Wk : Wq;
    float*       dst = isK ? Ks : Qs;
    const int    L   = isK ? LEN_K : LEN_Q;
    const int    mBase = (isK ? rowTile - 64 : rowTile) << 4;

    // A-row this lane feeds (row = mBase + (lane&15) for both lane halves)
    const int grow = mBase + lm;
    const float* arow;
    if (!isK) {
        arow = hmat + grow * D_MODEL;
    } else {
        arow = (grow < 1024) ? (mems + grow * D_MODEL)
                             : (hmat + (grow - 1024) * D_MODEL);
    }
    // B column = out-feature row of W (W is [out,in] so W^T[k][n] = W[n][k],
    // consecutive k are contiguous -> float2 loads)
    const float* brow = W + (nBase + lm) * D_MODEL;

    v8f acc = {};
    #pragma unroll 8
    for (int k0 = 0; k0 < D_MODEL; k0 += 4) {
        v2f a, b;
        a.x = arow[k0 + khalf];
        a.y = arow[k0 + khalf + 1];
        b.x = brow[k0 + khalf];
        b.y = brow[k0 + khalf + 1];
        acc = __builtin_amdgcn_wmma_f32_16x16x4_f32(
                  /*neg_a=*/false, a, /*neg_b=*/false, b,
                  /*c_mod=*/(short)0, acc,
                  /*reuse_a=*/false, /*reuse_b=*/false);
    }

    // Scale by paramR[head] and scatter to [H][B][L][16] layout.
    const int f  = nBase + lm;       // out feature; nBase%16==0 -> head uniform
    const int hh = f >> 4;
    const int d  = f & 15;
    const float r = paramR[hh];
    const int mOff = (lane < 16) ? 0 : 8;

    #pragma unroll
    for (int j = 0; j < 8; ++j) {
        const int row = mBase + mOff + j;     // row = l*BATCH + b
        const int l = row >> 2;
        const int b = row & 3;
        dst[(((hh * BATCH + b) * L) + l) * D_HEAD + d] = acc[j] * r;
    }
}

// ---------------------------------------------------------------------------
// Phase 2: score[q,k,b,h] = | prod_d sin(t)/t |, t = Qs[h,b,q,d] - Ks[h,b,k,d]
// 256 threads: tid&31 = b*H+h  (-> consecutive output floats, coalesced 128B
// stores per wave), tid>>5 = k sub-index (8 k's per block).
// Each thread keeps its K row in registers and streams 32 Q rows.
// Grid: x = Lk/8 = 64, y = Lq/32 = 8.
// ---------------------------------------------------------------------------
__device__ __forceinline__ float sinc_term(float qv, float kv) {
    const float t = qv - kv;
    return (t != 0.0f) ? __fdividef(__sinf(t), t) : 1.0f;
}

__device__ __forceinline__ float sinc_prod4(float4 a, float4 b) {
    float p = sinc_term(a.x, b.x);
    p *= sinc_term(a.y, b.y);
    p *= sinc_term(a.z, b.z);
    p *= sinc_term(a.w, b.w);
    return p;
}

__global__ __launch_bounds__(256)
void sinc_prod(const float* __restrict__ Qs,   // [H][B][Lq][16]
               const float* __restrict__ Ks,   // [H][B][Lk][16]
               float* __restrict__ out)        // [Lq][Lk][B][H]
{
    const int tid = threadIdx.x;
    const int bh  = tid & 31;            // b*N_HEAD + h (output inner index)
    const int kl  = tid >> 5;            // 0..7
    const int b   = bh >> 3;
    const int h   = bh & 7;
    const int hb  = h * BATCH + b;       // Qs/Ks major index

    const int k = (blockIdx.x << 3) + kl;

    const float4* krow = (const float4*)(Ks + (hb * LEN_K + k) * D_HEAD);
    const float4 k0 = krow[0], k1 = krow[1], k2 = krow[2], k3 = krow[3];

    const float4* qbase = (const float4*)(Qs + (hb * LEN_Q) * D_HEAD);
    const int q0 = blockIdx.y << 5;

    #pragma unroll 4
    for (int qi = 0; qi < 32; ++qi) {
        const int q = q0 + qi;
        const float4* qrow = qbase + q * 4;
        // prefetch next q row (gfx1250 global_prefetch_b8)
        if (qi + 1 < 32) __builtin_prefetch((const void*)(qrow + 4), 0, 3);

        const float4 a0 = qrow[0], a1 = qrow[1], a2 = qrow[2], a3 = qrow[3];

        float p = sinc_prod4(a0, k0);
        p *= sinc_prod4(a1, k1);
        p *= sinc_prod4(a2, k2);
        p *= sinc_prod4(a3, k3);

        out[(q * LEN_K + k) * (BATCH * N_HEAD) + bh] = __builtin_fabsf(p);
    }
}

// ---------------------------------------------------------------------------
extern "C" void kernel_launch(void* const* d_in, const int* in_sizes, int n_in,
                              void* d_out, int out_size, void* d_ws, size_t ws_size,
                              hipStream_t stream) {
    (void)in_sizes; (void)n_in; (void)out_size; (void)ws_size;

    const float* hmat   = (const float*)d_in[0];  // [256,4,128]
    const float* mems   = (const float*)d_in[1];  // [256,4,128]
    const float* Wq     = (const float*)d_in[2];  // [128,128]
    const float* Wk     = (const float*)d_in[3];  // [128,128]
    const float* paramR = (const float*)d_in[4];  // [8]
    float*       out    = (float*)d_out;          // [256,512,4,8]

    float* Qs = (float*)d_ws;                              // 512 KB
    float* Ks = Qs + (size_t)N_HEAD * BATCH * LEN_Q * D_HEAD; // 1 MB

    // Phase 1: 64 Q row-tiles + 128 K row-tiles, 8 feature tiles, 1 wave each
    dim3 g1(192, 8);
    proj_wmma<<<g1, 32, 0, stream>>>(hmat, mems, Wq, Wk, paramR, Qs, Ks);

    // Phase 2: 64 k-chunks x 8 q-chunks
    dim3 g2(LEN_K / 8, LEN_Q / 32);
    sinc_prod<<<g2, 256, 0, stream>>>(Qs, Ks, out);
}